// HybridMischiefModel_10771777978428
// MI455X (gfx1250) — compile-verified
//
#include <hip/hip_runtime.h>
#include <hip/hip_bf16.h>
#include <cstdint>

typedef float v2f __attribute__((ext_vector_type(2)));
typedef float v8f __attribute__((ext_vector_type(8)));

#define NN 65536
#define EE 1048576
#define BB 512

// ---------- helpers ----------
__device__ __forceinline__ unsigned fkey(float f) {
    unsigned u = __float_as_uint(f);
    return (u & 0x80000000u) ? ~u : (u | 0x80000000u);
}
__device__ __forceinline__ float funkey(unsigned k) {
    unsigned u = (k & 0x80000000u) ? (k & 0x7fffffffu) : ~k;
    return __uint_as_float(u);
}

// Async copy of 16 bytes global -> LDS (gfx1250 ASYNCcnt path)
__device__ __forceinline__ void async_g2l_b128(unsigned lds_off, const void* gaddr) {
    asm volatile("global_load_async_to_lds_b128 %0, %1, off"
                 :: "v"(lds_off), "v"((uint64_t)(uintptr_t)gaddr)
                 : "memory");
}
__device__ __forceinline__ void wait_asynccnt0() {
    asm volatile("s_wait_asynccnt 0x0" ::: "memory");
}

// ---------- generic GEMM: (rows x 256) @ (256 x 64), compile-time bias+relu ----------
// block = 256 threads = 8 waves; block covers 128 rows; wave w -> rows [w*16, w*16+16)
// each wave: 4 col-tiles of 16 -> full 64 output cols, K loop in steps of 4 (f32 WMMA)
template <bool RELU, bool HASBIAS>
__global__ void gemm_k256_n64(const float* __restrict__ A, const float* __restrict__ W,
                              const float* __restrict__ bias, float* __restrict__ out) {
    extern __shared__ float lds[];      // 256*64 floats = 64KB
    const int tid = threadIdx.x;

    // stage W via async global->LDS (16B per lane per issue, 16 issues = 64KB)
    {
        const unsigned base = (unsigned)(uintptr_t)&lds[0];
#pragma unroll
        for (int it = 0; it < 16; ++it) {
            const int i = tid + it * 256;               // float4 index, i < 4096
            async_g2l_b128(base + (unsigned)i * 16u, (const char*)W + (size_t)i * 16u);
        }
        wait_asynccnt0();
    }
    __syncthreads();

    const int wave = tid >> 5, lane = tid & 31;
    const int half = lane >> 4, l = lane & 15;
    const long rowBase = (long)blockIdx.x * 128 + wave * 16;
    const float2* A2 = (const float2*)A;
    const long arow = (rowBase + l) * 256;

    v8f acc[4] = {};
    for (int k = 0; k < 256; k += 4) {
        const int koff = k + half * 2;
        float2 af = A2[(arow + koff) >> 1];
        v2f a; a.x = af.x; a.y = af.y;
#pragma unroll
        for (int ct = 0; ct < 4; ++ct) {
            const int col = ct * 16 + l;
            v2f b; b.x = lds[koff * 64 + col]; b.y = lds[(koff + 1) * 64 + col];
            acc[ct] = __builtin_amdgcn_wmma_f32_16x16x4_f32(
                false, a, false, b, (short)0, acc[ct], false, false);
        }
    }
#pragma unroll
    for (int ct = 0; ct < 4; ++ct) {
        const int col = ct * 16 + l;
        const float bv = HASBIAS ? bias[col] : 0.f;
#pragma unroll
        for (int r = 0; r < 8; ++r) {
            const long row = rowBase + r + half * 8;
            float v = acc[ct][r] + bv;
            if (RELU) v = fmaxf(v, 0.f);
            out[row * 64 + col] = v;
        }
    }
}

// ---------- gat1 input GEMM: concat(x[13], vis[64]) (K=77 padded to 80) @ W(77x256) ----------
// grid.x covers 128-row blocks, grid.y = 4 col-blocks of 64
#define KP 80
#define AST 84
__global__ void gemm_gat1(const float* __restrict__ x13, const float* __restrict__ vis,
                          const float* __restrict__ W, float* __restrict__ h1) {
    extern __shared__ float lds[];             // 128*AST (A) + 80*64 (W) floats
    float* ldsA = lds;
    float* ldsW = lds + 128 * AST;
    const int tid = threadIdx.x;
    const long rowBase = (long)blockIdx.x * 128;
    const int colBase = blockIdx.y * 64;

    for (int i = tid; i < 128 * KP; i += 256) {
        const int r = i / KP, c = i - r * KP;
        const long row = rowBase + r;
        float v;
        if (c < 13)      v = x13[row * 13 + c];
        else if (c < 77) v = vis[row * 64 + (c - 13)];
        else             v = 0.f;
        ldsA[r * AST + c] = v;
    }
    for (int i = tid; i < KP * 64; i += 256) {
        const int k = i >> 6, c = i & 63;
        ldsW[i] = (k < 77) ? W[k * 256 + colBase + c] : 0.f;
    }
    __syncthreads();

    const int wave = tid >> 5, lane = tid & 31;
    const int half = lane >> 4, l = lane & 15;
    const int lr = wave * 16 + l;

    v8f acc[4] = {};
    for (int k = 0; k < KP; k += 4) {
        const int koff = k + half * 2;
        v2f a; a.x = ldsA[lr * AST + koff]; a.y = ldsA[lr * AST + koff + 1];
#pragma unroll
        for (int ct = 0; ct < 4; ++ct) {
            const int col = ct * 16 + l;
            v2f b; b.x = ldsW[koff * 64 + col]; b.y = ldsW[(koff + 1) * 64 + col];
            acc[ct] = __builtin_amdgcn_wmma_f32_16x16x4_f32(
                false, a, false, b, (short)0, acc[ct], false, false);
        }
    }
#pragma unroll
    for (int ct = 0; ct < 4; ++ct) {
        const int col = colBase + ct * 16 + l;
#pragma unroll
        for (int r = 0; r < 8; ++r) {
            const long row = rowBase + wave * 16 + r + half * 8;
            h1[row * 256 + col] = acc[ct][r];
        }
    }
}

// ---------- per-node attention coefficients a_src, a_dst ----------
template <int HH>
__global__ void node_attn(const float* __restrict__ h, const float* __restrict__ attS,
                          const float* __restrict__ attD, float* __restrict__ aS,
                          float* __restrict__ aD) {
    const long n = (long)blockIdx.x * blockDim.x + threadIdx.x;
    const float* row = h + n * (long)(HH * 64);
#pragma unroll
    for (int hh = 0; hh < HH; ++hh) {
        float s = 0.f, d = 0.f;
        for (int c = 0; c < 64; ++c) {
            const float v = row[hh * 64 + c];
            s += v * attS[hh * 64 + c];
            d += v * attD[hh * 64 + c];
        }
        aS[n * HH + hh] = s;
        aD[n * HH + hh] = d;
    }
}

// ---------- fold edge weights: fold1[k,h] = sum_c We1[k, h*64+c]*ae1[h,c]; fold2[k] likewise ----------
__global__ void fold_kernel(const float* __restrict__ We1, const float* __restrict__ ae1,
                            const float* __restrict__ We2, const float* __restrict__ ae2,
                            float* __restrict__ fold1, float* __restrict__ fold2) {
    const int t = threadIdx.x;
    if (t < 20) {
        const int k = t >> 2, hh = t & 3;
        float s = 0.f;
        for (int c = 0; c < 64; ++c) s += We1[k * 256 + hh * 64 + c] * ae1[hh * 64 + c];
        fold1[k * 4 + hh] = s;
    } else if (t < 25) {
        const int k = t - 20;
        float s = 0.f;
        for (int c = 0; c < 64; ++c) s += We2[k * 64 + c] * ae2[c];
        fold2[k] = s;
    }
}

// ---------- edge pass 1: leaky-relu logits + per-dst atomic max (uint-key) ----------
template <int HH>
__global__ void edge_logits(const int* __restrict__ src, const int* __restrict__ dst,
                            const float* __restrict__ ea, const float* __restrict__ aS,
                            const float* __restrict__ aD, const float* __restrict__ fold,
                            float* __restrict__ elog, unsigned* __restrict__ mkey) {
    const long e = (long)blockIdx.x * blockDim.x + threadIdx.x;
    const int s = src[e], d = dst[e];
    float a[5];
#pragma unroll
    for (int k = 0; k < 5; ++k) a[k] = ea[e * 5 + k];
#pragma unroll
    for (int hh = 0; hh < HH; ++hh) {
        float lg = aS[(long)s * HH + hh] + aD[(long)d * HH + hh];
#pragma unroll
        for (int k = 0; k < 5; ++k) lg += a[k] * fold[k * HH + hh];
        lg = (lg > 0.f) ? lg : 0.2f * lg;
        elog[e * HH + hh] = lg;
        atomicMax(&mkey[(long)d * HH + hh], fkey(lg));
    }
}

// ---------- edge pass 2: exp(logit - max) + per-dst atomic sum ----------
template <int HH>
__global__ void edge_exp(const int* __restrict__ dst, float* __restrict__ elog,
                         const unsigned* __restrict__ mkey, float* __restrict__ ssum) {
    const long e = (long)blockIdx.x * blockDim.x + threadIdx.x;
    const int d = dst[e];
#pragma unroll
    for (int hh = 0; hh < HH; ++hh) {
        float m = funkey(mkey[(long)d * HH + hh]);
        if (!__builtin_isfinite(m)) m = 0.f;
        const float ex = expf(elog[e * HH + hh] - m);
        elog[e * HH + hh] = ex;
        atomicAdd(&ssum[(long)d * HH + hh], ex);
    }
}

// ---------- aggregation gat1: wave per edge, 8 channels per lane (H=4, C=64) ----------
__global__ void agg256(const int* __restrict__ src, const int* __restrict__ dst,
                       const float* __restrict__ h, const float* __restrict__ elog,
                       const float* __restrict__ ssum, float* __restrict__ out) {
    const int wave = threadIdx.x >> 5, lane = threadIdx.x & 31;
    const long e = (long)blockIdx.x * 8 + wave;
    const int s = src[e], d = dst[e];
    const int hh = lane >> 3;
    const float w = elog[e * 4 + hh] / (ssum[(long)d * 4 + hh] + 1e-16f);
    const float4* hp = (const float4*)(h + (long)s * 256 + lane * 8);
    const float4 v0 = hp[0], v1 = hp[1];
    float* op = out + (long)d * 256 + lane * 8;
    atomicAdd(op + 0, w * v0.x); atomicAdd(op + 1, w * v0.y);
    atomicAdd(op + 2, w * v0.z); atomicAdd(op + 3, w * v0.w);
    atomicAdd(op + 4, w * v1.x); atomicAdd(op + 5, w * v1.y);
    atomicAdd(op + 6, w * v1.z); atomicAdd(op + 7, w * v1.w);
}

// ---------- aggregation gat2: wave per edge, 2 channels per lane (H=1, C=64) ----------
__global__ void agg64(const int* __restrict__ src, const int* __restrict__ dst,
                      const float* __restrict__ h, const float* __restrict__ elog,
                      const float* __restrict__ ssum, float* __restrict__ out) {
    const int wave = threadIdx.x >> 5, lane = threadIdx.x & 31;
    const long e = (long)blockIdx.x * 8 + wave;
    const int s = src[e], d = dst[e];
    const float w = elog[e] / (ssum[d] + 1e-16f);
    const float2 v = *(const float2*)(h + (long)s * 64 + lane * 2);
    float* op = out + (long)d * 64 + lane * 2;
    atomicAdd(op + 0, w * v.x);
    atomicAdd(op + 1, w * v.y);
}

// ---------- bias (+ optional relu), in place ----------
template <bool RELU, int NF>
__global__ void bias_act(float* __restrict__ buf, const float* __restrict__ bias) {
    const long i = (long)blockIdx.x * blockDim.x + threadIdx.x;
    float v = buf[i] + bias[(int)(i & (NF - 1))];
    if (RELU) v = fmaxf(v, 0.f);
    buf[i] = v;
}

// ---------- scene projection (512 x 512) @ (512 x 128) ----------
__global__ void scene_proj(const float* __restrict__ gf, const float* __restrict__ W,
                           const float* __restrict__ b, float* __restrict__ scene) {
    const int bi = blockIdx.x, c = threadIdx.x;
    float acc = b[c];
    for (int k = 0; k < 512; ++k) acc += gf[bi * 512 + k] * W[k * 128 + c];
    scene[bi * 128 + c] = acc;
}

// ---------- pooling gate ----------
__global__ void gate_logit(const float* __restrict__ h2, const float* __restrict__ gW,
                           const float* __restrict__ gb, const int* __restrict__ bv,
                           float* __restrict__ gl, unsigned* __restrict__ gmax) {
    const long n = (long)blockIdx.x * blockDim.x + threadIdx.x;
    float acc = gb[0];
    for (int c = 0; c < 64; ++c) acc += h2[n * 64 + c] * gW[c];
    gl[n] = acc;
    atomicMax(&gmax[bv[n]], fkey(acc));
}

__global__ void gate_exp(const int* __restrict__ bv, float* __restrict__ gl,
                         const unsigned* __restrict__ gmax, float* __restrict__ gsum) {
    const long n = (long)blockIdx.x * blockDim.x + threadIdx.x;
    const int b = bv[n];
    float m = funkey(gmax[b]);
    if (!__builtin_isfinite(m)) m = 0.f;
    const float ex = expf(gl[n] - m);
    gl[n] = ex;
    atomicAdd(&gsum[b], ex);
}

__global__ void pool_accum(const int* __restrict__ bv, const float* __restrict__ gl,
                           const float* __restrict__ gsum, const float* __restrict__ h2,
                           float* __restrict__ rel) {
    const long i = (long)blockIdx.x * blockDim.x + threadIdx.x;
    const long n = i >> 6;
    const int c = (int)(i & 63);
    const int b = bv[n];
    const float w = gl[n] / (gsum[b] + 1e-16f);
    atomicAdd(&rel[(long)b * 64 + c], w * h2[n * 64 + c]);
}

// ---------- classifier: [scene|rel](512x192) @ (192x64) relu @ (64x3) ----------
__global__ void classifier(const float* __restrict__ scene, const float* __restrict__ rel,
                           const float* __restrict__ W1, const float* __restrict__ b1,
                           const float* __restrict__ W2, const float* __restrict__ b2,
                           float* __restrict__ out) {
    __shared__ float hid[64];
    const int b = blockIdx.x, t = threadIdx.x;
    float acc = b1[t];
    for (int k = 0; k < 128; ++k) acc += scene[b * 128 + k] * W1[k * 64 + t];
    for (int k = 0; k < 64; ++k)  acc += rel[b * 64 + k] * W1[(128 + k) * 64 + t];
    hid[t] = fmaxf(acc, 0.f);
    __syncthreads();
    if (t < 3) {
        float o = b2[t];
        for (int k = 0; k < 64; ++k) o += hid[k] * W2[k * 3 + t];
        out[b * 3 + t] = o;
    }
}

extern "C" void kernel_launch(void* const* d_in, const int* in_sizes, int n_in,
                              void* d_out, int out_size, void* d_ws, size_t ws_size,
                              hipStream_t stream) {
    const float* gf      = (const float*)d_in[0];
    const float* x13     = (const float*)d_in[1];
    const float* roi     = (const float*)d_in[2];
    const int*   ei      = (const int*)d_in[3];
    const float* ea      = (const float*)d_in[4];
    const int*   bv      = (const int*)d_in[5];
    const float* roi_W   = (const float*)d_in[6];
    const float* roi_b   = (const float*)d_in[7];
    const float* scene_W = (const float*)d_in[8];
    const float* scene_b = (const float*)d_in[9];
    const float* g1_W    = (const float*)d_in[10];
    const float* g1_as   = (const float*)d_in[11];
    const float* g1_ad   = (const float*)d_in[12];
    const float* g1_We   = (const float*)d_in[13];
    const float* g1_ae   = (const float*)d_in[14];
    const float* g1_b    = (const float*)d_in[15];
    const float* g2_W    = (const float*)d_in[16];
    const float* g2_as   = (const float*)d_in[17];
    const float* g2_ad   = (const float*)d_in[18];
    const float* g2_We   = (const float*)d_in[19];
    const float* g2_ae   = (const float*)d_in[20];
    const float* g2_b    = (const float*)d_in[21];
    const float* gate_W  = (const float*)d_in[22];
    const float* gate_b  = (const float*)d_in[23];
    const float* cls1_W  = (const float*)d_in[24];
    const float* cls1_b  = (const float*)d_in[25];
    const float* cls2_W  = (const float*)d_in[26];
    const float* cls2_b  = (const float*)d_in[27];
    const int* src = ei;
    const int* dst = ei + EE;

    // workspace layout (floats)
    float* wsf = (float*)d_ws;
    size_t o = 0;
    float* vis  = wsf + o; o += (size_t)NN * 64;    // later reused as h2
    float* h1   = wsf + o; o += (size_t)NN * 256;   // later reused: out2 = h1, gatel = h1 + NN*64
    float* as1  = wsf + o; o += (size_t)NN * 4;
    float* ad1  = wsf + o; o += (size_t)NN * 4;
    float* e1   = wsf + o; o += (size_t)EE * 4;     // later reused as e2
    unsigned* m1 = (unsigned*)(wsf + o); o += (size_t)NN * 4;
    float* s1   = wsf + o; o += (size_t)NN * 4;
    float* out1 = wsf + o; o += (size_t)NN * 256;
    float* scn  = wsf + o; o += (size_t)BB * 128;
    float* rel  = wsf + o; o += (size_t)BB * 64;
    unsigned* gmax = (unsigned*)(wsf + o); o += BB;
    float* gsum = wsf + o; o += BB;
    float* fold1 = wsf + o; o += 32;
    float* fold2 = wsf + o; o += 8;
    float* h2    = vis;                 // safe: vis dead after gemm_gat1
    float* out2  = h1;                  // safe: h1 dead after agg256
    float* gatel = h1 + (size_t)NN * 64;

    // ---- phase 0: tiny precomputes ----
    fold_kernel<<<1, 32, 0, stream>>>(g1_We, g1_ae, g2_We, g2_ae, fold1, fold2);
    scene_proj<<<BB, 128, 0, stream>>>(gf, scene_W, scene_b, scn);

    // ---- phase 1: vis = relu(roi @ roi_W + roi_b)  [WMMA f32, async W stage] ----
    gemm_k256_n64<true, true><<<NN / 128, 256, 256 * 64 * 4, stream>>>(roi, roi_W, roi_b, vis);

    // ---- phase 2: h1 = concat(x, vis) @ g1_W  [WMMA f32, K padded 77->80] ----
    gemm_gat1<<<dim3(NN / 128, 4), 256, (128 * AST + KP * 64) * 4, stream>>>(x13, vis, g1_W, h1);

    // ---- gat1 attention ----
    node_attn<4><<<NN / 256, 256, 0, stream>>>(h1, g1_as, g1_ad, as1, ad1);
    hipMemsetAsync(m1, 0, (size_t)NN * 4 * sizeof(unsigned), stream);
    edge_logits<4><<<EE / 256, 256, 0, stream>>>(src, dst, ea, as1, ad1, fold1, e1, m1);
    hipMemsetAsync(s1, 0, (size_t)NN * 4 * sizeof(float), stream);
    edge_exp<4><<<EE / 256, 256, 0, stream>>>(dst, e1, m1, s1);
    hipMemsetAsync(out1, 0, (size_t)NN * 256 * sizeof(float), stream);
    agg256<<<EE / 8, 256, 0, stream>>>(src, dst, h1, e1, s1, out1);
    bias_act<true, 256><<<(NN * 256) / 256, 256, 0, stream>>>(out1, g1_b);

    // ---- gat2 ----
    gemm_k256_n64<false, false><<<NN / 128, 256, 256 * 64 * 4, stream>>>(out1, g2_W, nullptr, h2);
    node_attn<1><<<NN / 256, 256, 0, stream>>>(h2, g2_as, g2_ad, as1, ad1);
    hipMemsetAsync(m1, 0, (size_t)NN * sizeof(unsigned), stream);
    edge_logits<1><<<EE / 256, 256, 0, stream>>>(src, dst, ea, as1, ad1, fold2, e1, m1);
    hipMemsetAsync(s1, 0, (size_t)NN * sizeof(float), stream);
    edge_exp<1><<<EE / 256, 256, 0, stream>>>(dst, e1, m1, s1);
    hipMemsetAsync(out2, 0, (size_t)NN * 64 * sizeof(float), stream);
    agg64<<<EE / 8, 256, 0, stream>>>(src, dst, h2, e1, s1, out2);
    bias_act<false, 64><<<(NN * 64) / 256, 256, 0, stream>>>(out2, g2_b);

    // ---- global attention pooling ----
    hipMemsetAsync(gmax, 0, BB * sizeof(unsigned), stream);
    gate_logit<<<NN / 256, 256, 0, stream>>>(out2, gate_W, gate_b, bv, gatel, gmax);
    hipMemsetAsync(gsum, 0, BB * sizeof(float), stream);
    gate_exp<<<NN / 256, 256, 0, stream>>>(bv, gatel, gmax, gsum);
    hipMemsetAsync(rel, 0, (size_t)BB * 64 * sizeof(float), stream);
    pool_accum<<<(NN * 64) / 256, 256, 0, stream>>>(bv, gatel, gsum, out2, rel);

    // ---- classifier ----
    classifier<<<BB, 64, 0, stream>>>(scn, rel, cls1_W, cls1_b, cls2_W, cls2_b, (float*)d_out);
}